// LocalQKConv_25280177504269
// MI455X (gfx1250) — compile-verified
//
#include <hip/hip_runtime.h>
#include <cstdint>
#include <cstddef>

// ---------------------------------------------------------------------------
// LocalQKConv fused kernel for MI455X (gfx1250, wave32).
//
// Problem: B=4, N=8192, H=128, window W=3 sequence edges. No contraction
// anywhere -> WMMA inapplicable; this is an f32 VALU + LDS-locality problem.
// Strategy: tile (32 nodes x 32 channels) per 256-thread block, async-stage
// vec tile+halo into LDS (CDNA5 GLOBAL_LOAD_ASYNC_TO_LDS_B128 / ASYNCcnt),
// compute u in LDS, then fuse angle/dihedral/sigmoid/gather-sum in one pass.
// ---------------------------------------------------------------------------

namespace {

constexpr int Bc = 4;
constexpr int Nc = 8192;
constexpr int Hc = 128;
constexpr int Wc = 3;
constexpr float EPSc = 1e-8f;

constexpr int TN = 32;              // nodes per tile
constexpr int TH = 32;              // channels per tile (== wavefront lanes)
constexpr int HALO_V = 2 * Wc;      // vec halo needed to build u halo (6)
constexpr int HALO_U = Wc;          // u halo needed for u_j (3)
constexpr int NV = TN + 2 * HALO_V; // 44 staged vec nodes
constexpr int NU = TN + 2 * HALO_U; // 38 computed u nodes
constexpr int THREADS = 256;

} // namespace

// ---- CDNA5 async global->LDS copy (16B per lane), with asm fallback -------
typedef int v4i_t __attribute__((ext_vector_type(4)));
typedef __attribute__((address_space(1))) v4i_t global_v4i;
typedef __attribute__((address_space(3))) v4i_t shared_v4i;

__device__ __forceinline__ void async_copy16(const float* g, float* l) {
#if __has_builtin(__builtin_amdgcn_global_load_async_to_lds_b128)
  __builtin_amdgcn_global_load_async_to_lds_b128(
      (global_v4i*)g, (shared_v4i*)l, /*imm offset*/ 0, /*cpol*/ 0);
#else
  unsigned loff = (unsigned)(size_t)(__attribute__((address_space(3))) char*)l;
  asm volatile("global_load_async_to_lds_b128 %0, %1, off"
               :
               : "v"(loff), "v"(g)
               : "memory");
#endif
}

__device__ __forceinline__ void wait_async_zero() {
#if __has_builtin(__builtin_amdgcn_s_wait_asynccnt)
  __builtin_amdgcn_s_wait_asynccnt(0);
#else
  asm volatile("s_wait_asynccnt 0" ::: "memory");
#endif
}

__device__ __forceinline__ float clamp1(float x) {
  return fminf(fmaxf(x, -1.0f), 1.0f);
}

__device__ __forceinline__ float sigmoid_fast(float z) {
  // 1/(1+e^-z): v_exp_f32 + v_rcp_f32 (both ~1 ulp, fine for gate values)
  return __builtin_amdgcn_rcpf(1.0f + __expf(-z));
}

__global__ __launch_bounds__(THREADS)
void localqk_fused(const float* __restrict__ xs,   // (B,N,H)
                   const float* __restrict__ vec,  // (B,N,3,H)
                   const float* __restrict__ waq, const float* __restrict__ wdq,
                   const float* __restrict__ bq,
                   const float* __restrict__ wak, const float* __restrict__ wdk,
                   const float* __restrict__ bk,
                   float* __restrict__ qout,       // (B,N,H)
                   float* __restrict__ kout)       // (B,N,H)
{
  const int tid = threadIdx.x;
  const int n0  = blockIdx.x * TN;        // first node of tile
  const int h0  = blockIdx.y * TH;        // first channel of tile
  const int b   = blockIdx.z;
  const int hh  = tid & (TH - 1);         // channel lane within tile
  const int h   = h0 + hh;

  // LDS: vec tile+halo (44*3*32 f32 = 16.9KB) and u tile+halo (38*3*32 = 14.6KB)
  __shared__ float sv[NV * 3 * TH];
  __shared__ float su[NU * 3 * TH];

  const size_t vbase = (size_t)b * Nc * 3 * Hc;

  // ---- Phase 1: async-stage vec[n0-6 .. n0+TN+6) x 3 comps x TH channels ---
  // Each (node,comp) row = TH floats = 8 x b128 lane transfers.
  {
    const int total = NV * 3 * (TH / 4);  // 1056 lane transfers
    for (int idx = tid; idx < total; idx += THREADS) {
      const int row = idx >> 3;           // (node,comp) row, 0..131
      const int l4  = (idx & 7) * 4;      // float offset within row
      const int m   = n0 - HALO_V + row / 3;
      const int c   = row % 3;
      if (m >= 0 && m < Nc) {
        const float* gp = vec + vbase + ((size_t)m * 3 + c) * Hc + (h0 + l4);
        float* lp = &sv[row * TH + l4];
        async_copy16(gp, lp);
      }
    }
  }
  wait_async_zero();   // this wave's async LDS writes complete
  __syncthreads();     // visible to all waves in the workgroup

  // ---- Phase 2: u[m] = sum_d normalize(vec[m+d]-vec[m]), m in tile+-3 -----
  {
    const int total = NU * TH;            // 1216 tasks
    for (int idx = tid; idx < total; idx += THREADS) {
      const int ch = idx & (TH - 1);
      const int ml = idx >> 5;            // 0..NU-1
      const int m  = n0 - HALO_U + ml;
      float ux = 0.0f, uy = 0.0f, uz = 0.0f;
      if (m >= 0 && m < Nc) {
        const int vr = (ml + (HALO_V - HALO_U)) * 3;  // m's row in sv
        const float ax = sv[(vr + 0) * TH + ch];
        const float ay = sv[(vr + 1) * TH + ch];
        const float az = sv[(vr + 2) * TH + ch];
#pragma unroll
        for (int d = -Wc; d <= Wc; ++d) {
          if (d == 0) continue;
          const int j = m + d;
          if (j < 0 || j >= Nc) continue;
          const int jr = (ml + (HALO_V - HALO_U) + d) * 3;
          const float bx = sv[(jr + 0) * TH + ch] - ax;
          const float by = sv[(jr + 1) * TH + ch] - ay;
          const float bz = sv[(jr + 2) * TH + ch] - az;
          const float nb  = __builtin_sqrtf(bx * bx + by * by + bz * bz);
          const float inv = __builtin_amdgcn_rcpf(fmaxf(nb, EPSc));
          ux += bx * inv; uy += by * inv; uz += bz * inv;
        }
      }
      su[(ml * 3 + 0) * TH + ch] = ux;
      su[(ml * 3 + 1) * TH + ch] = uy;
      su[(ml * 3 + 2) * TH + ch] = uz;
    }
  }
  __syncthreads();

  // ---- Phase 3: per-(n,h) edge features + gated gather into q,k ----------
  const float vaq = waq[h], vdq = wdq[h], vbq = bq[h];
  const float vak = wak[h], vdk = wdk[h], vbk = bk[h];
  const size_t sbase = (size_t)b * Nc * Hc;
  const int nsub = tid >> 5;              // 0..7: node sub-group

#pragma unroll
  for (int k = 0; k < TN / 8; ++k) {      // 4 nodes per thread
    const int nl = nsub + k * 8;          // node within tile
    const int n  = n0 + nl;

    const int vr = (nl + HALO_V) * 3;
    const float ax = sv[(vr + 0) * TH + hh];
    const float ay = sv[(vr + 1) * TH + hh];
    const float az = sv[(vr + 2) * TH + hh];
    const int ur = (nl + HALO_U) * 3;
    const float ux = su[(ur + 0) * TH + hh];
    const float uy = su[(ur + 1) * TH + hh];
    const float uz = su[(ur + 2) * TH + hh];
    const float inv_nu = __builtin_amdgcn_rcpf(
        fmaxf(__builtin_sqrtf(ux * ux + uy * uy + uz * uz), EPSc));

    float qa = 0.0f, ka = 0.0f;
#pragma unroll
    for (int d = -Wc; d <= Wc; ++d) {
      if (d == 0) continue;
      const int j = n + d;
      if (j < 0 || j >= Nc) continue;

      // e_ij = normalize(v_j - v_i)   (recomputed from LDS; cheaper than
      // storing 300MB of per-edge e through HBM)
      const int jv = (nl + HALO_V + d) * 3;
      const float bx = sv[(jv + 0) * TH + hh] - ax;
      const float by = sv[(jv + 1) * TH + hh] - ay;
      const float bz = sv[(jv + 2) * TH + hh] - az;
      const float nb  = __builtin_sqrtf(bx * bx + by * by + bz * bz);
      const float inv = __builtin_amdgcn_rcpf(fmaxf(nb, EPSc));
      const float ex = bx * inv, ey = by * inv, ez = bz * inv;

      // angle feature
      const float due = ux * ex + uy * ey + uz * ez;
      const float ang = clamp1(due * inv_nu);

      // dihedral feature
      const float pix = ux - due * ex;
      const float piy = uy - due * ey;
      const float piz = uz - due * ez;
      const int ju = (nl + HALO_U + d) * 3;
      const float wx = su[(ju + 0) * TH + hh];
      const float wy = su[(ju + 1) * TH + hh];
      const float wz = su[(ju + 2) * TH + hh];
      const float duj = wx * ex + wy * ey + wz * ez;
      const float pjx = wx - duj * ex;
      const float pjy = wy - duj * ey;
      const float pjz = wz - duj * ez;
      const float dp  = pix * pjx + piy * pjy + piz * pjz;
      const float npi = fmaxf(__builtin_sqrtf(pix * pix + piy * piy + piz * piz), EPSc);
      const float npj = fmaxf(__builtin_sqrtf(pjx * pjx + pjy * pjy + pjz * pjz), EPSc);
      const float dih = clamp1(dp * __builtin_amdgcn_rcpf(npi * npj));

      // gates + gather (x_scalar read hits L1/L2: +-3 row locality)
      const float sj = xs[sbase + (size_t)j * Hc + h];
      const float gq = sigmoid_fast(ang * vaq + dih * vdq + vbq);
      const float gk = sigmoid_fast(ang * vak + dih * vdk + vbk);
      qa += gq * sj;
      ka += gk * sj;
    }

    const size_t o = sbase + (size_t)n * Hc + h;
    qout[o] = qa;
    kout[o] = ka;
  }
}

extern "C" void kernel_launch(void* const* d_in, const int* in_sizes, int n_in,
                              void* d_out, int out_size, void* d_ws, size_t ws_size,
                              hipStream_t stream) {
  // setup_inputs order:
  // 0 x_scalar (B,N,H) f32 | 1 vec (B,N,3,H) f32 | 2 src i32 | 3 dst i32
  // 4 w_angle_q | 5 w_dih_q | 6 b_q | 7 w_angle_k | 8 w_dih_k | 9 b_k  (H,) f32
  // src/dst encode the deterministic +-3 sequence window; regenerated on the fly.
  const float* xs  = (const float*)d_in[0];
  const float* vec = (const float*)d_in[1];
  const float* waq = (const float*)d_in[4];
  const float* wdq = (const float*)d_in[5];
  const float* bq  = (const float*)d_in[6];
  const float* wak = (const float*)d_in[7];
  const float* wdk = (const float*)d_in[8];
  const float* bk  = (const float*)d_in[9];

  float* qout = (float*)d_out;
  float* kout = qout + (size_t)Bc * Nc * Hc;

  dim3 grid(Nc / TN, Hc / TH, Bc);   // 256 x 4 x 4 blocks
  dim3 block(THREADS);               // 8 wave32 per block
  hipLaunchKernelGGL(localqk_fused, grid, block, 0, stream,
                     xs, vec, waq, wdq, bq, wak, wdk, bk, qout, kout);

  (void)in_sizes; (void)n_in; (void)d_ws; (void)ws_size; (void)out_size;
}